// dspo_model_83030307766874
// MI455X (gfx1250) — compile-verified
//
#include <hip/hip_runtime.h>
#include <hip/hip_bf16.h>

// ---------------------------------------------------------------------------
// Kalman-style linear recurrence as a 3-level blocked parallel scan.
//   x[k+1] = M x[k] + L y[:,k],  yhat[:,k] = H x[k],  M = A - L H
// Level 0: 64 chunks of 64 steps per workgroup, batched as 64x64 WMMA matmuls.
// Level 1: per-64-chunk-block affine scan (Phi = M^64).
// Level 2: 64 sequential block steps (PhiB = M^4096).
// fp32 WMMA (V_WMMA_F32_16X16X4_F32) keeps the 262144-step feedback at fp32.
//
// Schedule: M and L A-fragments pinned in VGPRs; all B-fragments of a step
// batch-loaded from LDS before the WMMA chains (one wait, clause-able loads);
// next step's observation gather prefetched into registers during compute.
// ---------------------------------------------------------------------------

typedef __attribute__((ext_vector_type(2))) float v2f;
typedef __attribute__((ext_vector_type(8))) float v8f;

#define NTOT 262144   // horizon
#define NS   64       // state dim
#define MO   32       // observation dim
#define TCH  64       // steps per chunk
#define CCH  4096     // chunks  (NTOT / TCH)
#define NBLK 64       // workgroups / chunk blocks (CCH / 64)
#define XLD  76       // padded LDS row stride (bank-conflict avoidance)

__device__ __forceinline__ v8f wmma4(v2f a, v2f b, v8f c) {
  // D = A(16x4) * B(4x16) + C, fp32 everywhere.
  return __builtin_amdgcn_wmma_f32_16x16x4_f32(
      /*neg_a=*/false, a, /*neg_b=*/false, b,
      /*c_mod=*/(short)0, c, /*reuse_a=*/false, /*reuse_b=*/false);
}

// A-matrix 16x4 fragment (ISA 7.12.2): lane l holds row rowbase+(l&15),
// VGPR0/1 = K = kbase + 2*(l>>4) + {0,1}.
__device__ __forceinline__ v2f afrag(const float* mat, int ld, int rowbase,
                                     int kbase, int lane) {
  int r = rowbase + (lane & 15);
  int k = kbase + ((lane >> 4) << 1);
  v2f a;
  a.x = mat[r * ld + k];
  a.y = mat[r * ld + k + 1];
  return a;
}

// B-matrix 4x16 fragment: lane l holds col nbase+(l&15),
// VGPR0/1 = K = kbase + 2*(l>>4) + {0,1}.
__device__ __forceinline__ v2f bfrag(const float* mat, int ld, int kbase,
                                     int nbase, int lane) {
  int k = kbase + ((lane >> 4) << 1);
  int n = nbase + (lane & 15);
  v2f b;
  b.x = mat[k * ld + n];
  b.y = mat[(k + 1) * ld + n];
  return b;
}

// ---------------------------------------------------------------------------
// prep: M = A - L*H; Phi = M^64; PhiB = M^4096 via 12 repeated squarings.
// ---------------------------------------------------------------------------
__global__ void prep_kernel(const float* __restrict__ A,
                            const float* __restrict__ H,
                            const float* __restrict__ L,
                            float* __restrict__ Mout,
                            float* __restrict__ Phi,
                            float* __restrict__ PhiB) {
  __shared__ float P[NS * NS];
  __shared__ float Q[NS * NS];
  for (int idx = threadIdx.x; idx < NS * NS; idx += blockDim.x) {
    int r = idx >> 6, c = idx & 63;
    float s = A[idx];
    for (int k = 0; k < MO; ++k) s -= L[r * MO + k] * H[k * NS + c];
    Mout[idx] = s;
    P[idx] = s;
  }
  __syncthreads();
  float* src = P;
  float* dst = Q;
  for (int sq = 0; sq < 12; ++sq) {   // dst = M^(2^(sq+1))
    for (int idx = threadIdx.x; idx < NS * NS; idx += blockDim.x) {
      int r = idx >> 6, c = idx & 63;
      float s = 0.f;
      for (int k = 0; k < NS; ++k) s += src[r * NS + k] * src[k * NS + c];
      dst[idx] = s;
    }
    __syncthreads();
    if (sq == 5)
      for (int idx = threadIdx.x; idx < NS * NS; idx += blockDim.x)
        Phi[idx] = dst[idx];          // M^64
    if (sq == 11)
      for (int idx = threadIdx.x; idx < NS * NS; idx += blockDim.x)
        PhiB[idx] = dst[idx];         // M^4096
    __syncthreads();
    float* t = src; src = dst; dst = t;
  }
}

// ---------------------------------------------------------------------------
// scan: 64 chunk columns per workgroup; 8 waves.
//   wave w: column-tile c = w&3, row-pair p = w>>2 (rows [32p,32p+32) of X').
// Per step: X' = M X + L Y_t (and Yhat = H X from the PRE-update state when
// WRITE_OUT). State double-buffered in LDS.
// ---------------------------------------------------------------------------
template <bool WRITE_OUT>
__global__ void __launch_bounds__(256)
scan_kernel(const float* __restrict__ y, const float* __restrict__ Mg,
            const float* __restrict__ Lg, const float* __restrict__ Hg,
            const float* __restrict__ Xstart, float* __restrict__ Vend,
            float* __restrict__ out) {
  __shared__ float Xs[2 * NS * XLD];   // double-buffered 64x64 state tile
  __shared__ float Yb[MO * XLD];       // this step's 32x64 observation tile
  __shared__ float Hsh[MO * 65];       // H, padded

  const int tid = threadIdx.x;
  const int lane = tid & 31;
  const int w = tid >> 5;
  const int c = w & 3;
  const int p = w >> 2;
  const int blk = blockIdx.x;

  if (WRITE_OUT) {
    for (int idx = tid; idx < MO * NS; idx += 256)
      Hsh[(idx >> 6) * 65 + (idx & 63)] = Hg[idx];
  }
  for (int idx = tid; idx < NS * NS; idx += 256) {
    int r = idx >> 6, j = idx & 63;
    Xs[r * XLD + j] = WRITE_OUT ? Xstart[(size_t)blk * 4096 + idx] : 0.f;
  }

  // Loop-invariant A-fragments of M and L pinned in VGPRs.
  v2f Ma[2][16];
  v2f La[2][8];
#pragma unroll
  for (int q = 0; q < 2; ++q) {
#pragma unroll
    for (int kk = 0; kk < 16; ++kk)
      Ma[q][kk] = afrag(Mg, NS, (p * 2 + q) * 16, kk * 4, lane);
#pragma unroll
    for (int kk = 0; kk < 8; ++kk)
      La[q][kk] = afrag(Lg, MO, (p * 2 + q) * 16, kk * 4, lane);
  }

  // Per-thread gather addresses for the observation tile:
  // element s covers Yb[i][j] with idx = tid + 256*s, i = idx>>6, j = idx&63.
  size_t ybase[8];
  int ylds[8];
#pragma unroll
  for (int s = 0; s < 8; ++s) {
    int idx = tid + s * 256;
    int i = idx >> 6, j = idx & 63;
    ybase[s] = (size_t)i * NTOT + (size_t)(blk * 64 + j) * TCH;
    ylds[s] = i * XLD + j;
  }

  // Prologue: fetch step-0 observations into registers.
  float yv[8];
#pragma unroll
  for (int s = 0; s < 8; ++s) yv[s] = y[ybase[s]];

  __syncthreads();

  int buf = 0;
  for (int t = 0; t < TCH; ++t) {
    // Commit this step's observations to LDS.
#pragma unroll
    for (int s = 0; s < 8; ++s) Yb[ylds[s]] = yv[s];
    __syncthreads();

    // Prefetch next step's observations (global latency hides under WMMA).
    if (t + 1 < TCH) {
#pragma unroll
      for (int s = 0; s < 8; ++s) yv[s] = y[ybase[s] + (t + 1)];
    }

    const float* Xcur = &Xs[buf * NS * XLD];

    // Batch-load all B-fragments for this step (clause-able ds loads, one
    // wait before the WMMA chains instead of one wait per fragment).
    v2f Bx[16];
#pragma unroll
    for (int kk = 0; kk < 16; ++kk)
      Bx[kk] = bfrag(Xcur, XLD, kk * 4, c * 16, lane);
    v2f By[8];
#pragma unroll
    for (int kk = 0; kk < 8; ++kk)
      By[kk] = bfrag(Yb, XLD, kk * 4, c * 16, lane);
    v2f Ha[16];
    if (WRITE_OUT) {
#pragma unroll
      for (int kk = 0; kk < 16; ++kk)
        Ha[kk] = afrag(Hsh, 65, p * 16, kk * 4, lane);
    }

    v8f acc0 = {};   // rows [32p, 32p+16) of X'
    v8f acc1 = {};   // rows [32p+16, 32p+32) of X'
    v8f acch = {};   // rows [16p, 16p+16) of Yhat
#pragma unroll
    for (int kk = 0; kk < 16; ++kk) {          // M*X (+ H*X) over K=64
      acc0 = wmma4(Ma[0][kk], Bx[kk], acc0);
      acc1 = wmma4(Ma[1][kk], Bx[kk], acc1);
      if (WRITE_OUT) acch = wmma4(Ha[kk], Bx[kk], acch);
    }
#pragma unroll
    for (int kk = 0; kk < 8; ++kk) {           // + L*Y over K=32
      acc0 = wmma4(La[0][kk], By[kk], acc0);
      acc1 = wmma4(La[1][kk], By[kk], acc1);
    }

    float* Xnxt = &Xs[(buf ^ 1) * NS * XLD];
    const int colj = c * 16 + (lane & 15);
    const int rhal = (lane >> 4) << 3;   // +8 rows for upper lane half
#pragma unroll
    for (int v = 0; v < 8; ++v) {
      Xnxt[((p * 2) * 16 + v + rhal) * XLD + colj] = acc0[v];
      Xnxt[((p * 2 + 1) * 16 + v + rhal) * XLD + colj] = acc1[v];
    }
    if (WRITE_OUT) {
#pragma unroll
      for (int v = 0; v < 8; ++v) {
        int i = p * 16 + v + rhal;
        out[(size_t)i * NTOT + (size_t)(blk * 64 + colj) * TCH + t] = acch[v];
      }
    }
    __syncthreads();
    buf ^= 1;
  }

  if (!WRITE_OUT) {   // dump zero-start chunk end states v_c
    const float* Xf = &Xs[buf * NS * XLD];
    for (int idx = tid; idx < NS * NS; idx += 256)
      Vend[(size_t)blk * 4096 + idx] = Xf[(idx >> 6) * XLD + (idx & 63)];
  }
}

// ---------------------------------------------------------------------------
// Boundary-state propagation (3 tiny kernels of 64-dim affine matvec scans).
// ---------------------------------------------------------------------------
__global__ void blkend_kernel(const float* __restrict__ V,
                              const float* __restrict__ Phi,
                              float* __restrict__ BlkEnd) {
  __shared__ float xs[NS];
  const int r = threadIdx.x;
  const int b = blockIdx.x;
  float x = 0.f;
  for (int j = 0; j < 64; ++j) {
    xs[r] = x;
    __syncthreads();
    float s = V[(size_t)b * 4096 + r * 64 + j];
#pragma unroll 8
    for (int k = 0; k < NS; ++k) s += Phi[r * NS + k] * xs[k];
    __syncthreads();
    x = s;
  }
  BlkEnd[b * NS + r] = x;
}

__global__ void topscan_kernel(const float* __restrict__ BlkEnd,
                               const float* __restrict__ PhiB,
                               float* __restrict__ BlkStart) {
  __shared__ float xs[NS];
  const int r = threadIdx.x;
  float g = 0.f;
  for (int b = 0; b < NBLK; ++b) {
    BlkStart[b * NS + r] = g;
    xs[r] = g;
    __syncthreads();
    float s = BlkEnd[b * NS + r];
#pragma unroll 8
    for (int k = 0; k < NS; ++k) s += PhiB[r * NS + k] * xs[k];
    __syncthreads();
    g = s;
  }
}

__global__ void chunkstart_kernel(const float* __restrict__ V,
                                  const float* __restrict__ Phi,
                                  const float* __restrict__ BlkStart,
                                  float* __restrict__ Xstart) {
  __shared__ float xs[NS];
  const int r = threadIdx.x;
  const int b = blockIdx.x;
  float x = BlkStart[b * NS + r];
  for (int j = 0; j < 64; ++j) {
    Xstart[(size_t)b * 4096 + r * 64 + j] = x;   // state at chunk start
    xs[r] = x;
    __syncthreads();
    float s = V[(size_t)b * 4096 + r * 64 + j];
#pragma unroll 8
    for (int k = 0; k < NS; ++k) s += Phi[r * NS + k] * xs[k];
    __syncthreads();
    x = s;
  }
}

// ---------------------------------------------------------------------------
extern "C" void kernel_launch(void* const* d_in, const int* in_sizes, int n_in,
                              void* d_out, int out_size, void* d_ws,
                              size_t ws_size, hipStream_t stream) {
  (void)in_sizes; (void)n_in; (void)out_size; (void)ws_size;
  const float* y = (const float*)d_in[0];   // (32, 262144)
  const float* A = (const float*)d_in[1];   // (64, 64)
  const float* H = (const float*)d_in[2];   // (32, 64)
  const float* L = (const float*)d_in[3];   // (64, 32)
  float* out = (float*)d_out;               // (32, 262144)

  // Workspace layout (floats): ~2.2 MB total.
  float* ws = (float*)d_ws;
  float* wsM    = ws;                       // 4096
  float* wsPhi  = ws + 4096;                // 4096
  float* wsPhiB = ws + 8192;                // 4096
  float* wsV    = ws + 12288;               // CCH*64 = 262144
  float* wsXs   = wsV + (size_t)CCH * 64;   // 262144
  float* wsBE   = wsXs + (size_t)CCH * 64;  // NBLK*64
  float* wsBS   = wsBE + NBLK * NS;         // NBLK*64

  prep_kernel<<<1, 256, 0, stream>>>(A, H, L, wsM, wsPhi, wsPhiB);
  scan_kernel<false><<<NBLK, 256, 0, stream>>>(y, wsM, L, H, nullptr, wsV,
                                               nullptr);
  blkend_kernel<<<NBLK, NS, 0, stream>>>(wsV, wsPhi, wsBE);
  topscan_kernel<<<1, NS, 0, stream>>>(wsBE, wsPhiB, wsBS);
  chunkstart_kernel<<<NBLK, NS, 0, stream>>>(wsV, wsPhi, wsBS, wsXs);
  scan_kernel<true><<<NBLK, 256, 0, stream>>>(y, wsM, L, H, wsXs, nullptr,
                                              out);
}